// GCN_62843961475710
// MI455X (gfx1250) — compile-verified
//
#include <hip/hip_runtime.h>

typedef float v2f __attribute__((ext_vector_type(2)));
typedef float v8f __attribute__((ext_vector_type(8)));

#define WMMA_F32(a, b, c) \
    __builtin_amdgcn_wmma_f32_16x16x4_f32(false, (a), false, (b), (short)0, (c), false, false)

// ---------------------------------------------------------------------------
// Degree / norm kernels
// ---------------------------------------------------------------------------
__global__ void deg_init_kernel(float* __restrict__ deg, int n) {
    int t = blockIdx.x * blockDim.x + threadIdx.x;
    if (t < n) deg[t] = 1.0f;   // self-loop contributes 1 to every node
}

__global__ void deg_accum_kernel(const long long* __restrict__ dst,
                                 float* __restrict__ deg, long long E) {
    long long t = (long long)blockIdx.x * blockDim.x + threadIdx.x;
    if (t < E) atomicAdd(&deg[dst[t]], 1.0f);
}

__global__ void deg_rsqrt_kernel(float* __restrict__ deg, int n) {
    int t = blockIdx.x * blockDim.x + threadIdx.x;
    if (t < n) {
        float d = deg[t];
        deg[t] = (d > 0.0f) ? rsqrtf(d) : 0.0f;
    }
}

// ---------------------------------------------------------------------------
// GEMM1: h1[N,64] = x[N,128] @ W1[128,64]   (V_WMMA_F32_16X16X4_F32)
// One wave handles TWO 16-row M tiles (32 rows) and all 64 output columns:
// 8 C-fragments; each B fragment loaded from LDS is reused for 2 WMMAs.
// W1 staged in LDS, swizzled so (k, k+1) pairs per column are contiguous
// (single ds_load_b64 per B fragment).
// ---------------------------------------------------------------------------
__global__ __launch_bounds__(256)
void gemm1_kernel(const float* __restrict__ x, const float* __restrict__ W1,
                  float* __restrict__ h1, int n_nodes) {
    __shared__ float sB[128 * 64];   // swizzled: sB[(k>>1)*128 + n*2 + (k&1)]

    for (int i = threadIdx.x; i < 128 * 64; i += 256) {
        int k = i >> 6;          // i / 64
        int n = i & 63;          // i % 64
        sB[(k >> 1) * 128 + n * 2 + (k & 1)] = W1[i];
    }
    __syncthreads();

    const int wave = threadIdx.x >> 5;
    const int lane = threadIdx.x & 31;
    const int half = lane >> 4;      // 0: lanes 0-15, 1: lanes 16-31
    const int l16  = lane & 15;
    const long long base = ((long long)blockIdx.x * 8 + wave) * 32;

    if (base >= n_nodes) return;     // wave-uniform exit (EXEC stays full)

    long long r0 = base + l16;       if (r0 >= n_nodes) r0 = n_nodes - 1;
    long long r1 = base + 16 + l16;  if (r1 >= n_nodes) r1 = n_nodes - 1;
    const float* a0p = x + r0 * 128;
    const float* a1p = x + r1 * 128;

    v8f c0[4] = {};
    v8f c1[4] = {};
    for (int kb = 0; kb < 128; kb += 4) {
        const int klo = kb + half * 2;                 // this lane's K pair
        v2f a0 = *(const v2f*)(a0p + klo);
        v2f a1 = *(const v2f*)(a1p + klo);
        const float* bbase = sB + (klo >> 1) * 128;
#pragma unroll
        for (int nt = 0; nt < 4; nt++) {
            v2f b = *(const v2f*)(bbase + (nt * 16 + l16) * 2);
            c0[nt] = WMMA_F32(a0, b, c0[nt]);
            c1[nt] = WMMA_F32(a1, b, c1[nt]);
        }
    }

    if (base + 32 <= n_nodes) {
        // full tile: unguarded coalesced stores
#pragma unroll
        for (int nt = 0; nt < 4; nt++) {
            int col = nt * 16 + l16;
#pragma unroll
            for (int j = 0; j < 8; j++) {
                long long rr = base + j + half * 8;
                h1[rr * 64 + col]        = c0[nt][j];
                h1[(rr + 16) * 64 + col] = c1[nt][j];
            }
        }
    } else {
#pragma unroll
        for (int nt = 0; nt < 4; nt++) {
            int col = nt * 16 + l16;
#pragma unroll
            for (int j = 0; j < 8; j++) {
                long long rr = base + j + half * 8;
                if (rr < n_nodes)      h1[rr * 64 + col]        = c0[nt][j];
                if (rr + 16 < n_nodes) h1[(rr + 16) * 64 + col] = c1[nt][j];
            }
        }
    }
}

// ---------------------------------------------------------------------------
// Edge scatter, layer 1: agg[dst] += h1[src] * norm   (64 floats per edge,
// 16 lanes/edge, float4 per lane; self loops appended at e >= E)
// ---------------------------------------------------------------------------
__global__ void scatter64_kernel(const float* __restrict__ h,
                                 const long long* __restrict__ src,
                                 const long long* __restrict__ dst,
                                 const float* __restrict__ dinv,
                                 float* __restrict__ agg,
                                 long long E, int n_nodes) {
    long long t = (long long)blockIdx.x * blockDim.x + threadIdx.x;
    long long e = t >> 4;
    if (e >= E + n_nodes) return;
    int seg = (int)(t & 15) * 4;

    long long s, d;
    float w;
    if (e < E) {
        s = src[e]; d = dst[e];
        w = dinv[s] * dinv[d];
    } else {
        s = d = e - E;                 // self loop
        float di = dinv[s];
        w = di * di;
    }
    const float4 v = *(const float4*)(h + s * 64 + seg);
    float* o = agg + d * 64 + seg;
    atomicAdd(o + 0, v.x * w);
    atomicAdd(o + 1, v.y * w);
    atomicAdd(o + 2, v.z * w);
    atomicAdd(o + 3, v.w * w);
}

// ---------------------------------------------------------------------------
// hidden = relu(agg + b1), in place (64 cols)
// ---------------------------------------------------------------------------
__global__ void bias_relu_kernel(float* __restrict__ h, const float* __restrict__ b,
                                 long long total) {
    long long t = (long long)blockIdx.x * blockDim.x + threadIdx.x;
    if (t < total) h[t] = fmaxf(h[t] + b[t & 63], 0.0f);
}

// ---------------------------------------------------------------------------
// GEMM2: h2[N,40] = hidden[N,64] @ W2[64,40]   (N padded to 48 => 3 C-frags,
// two 16-row M tiles per wave)
// ---------------------------------------------------------------------------
__global__ __launch_bounds__(256)
void gemm2_kernel(const float* __restrict__ hid, const float* __restrict__ W2,
                  float* __restrict__ h2, int n_nodes) {
    __shared__ float sB[64 * 48];    // swizzled padded: sB[(k>>1)*96 + n*2 + (k&1)]

    for (int i = threadIdx.x; i < 64 * 48; i += 256) {
        int k = i / 48;
        int n = i % 48;
        float v = (n < 40) ? W2[k * 40 + n] : 0.0f;
        sB[(k >> 1) * 96 + n * 2 + (k & 1)] = v;
    }
    __syncthreads();

    const int wave = threadIdx.x >> 5;
    const int lane = threadIdx.x & 31;
    const int half = lane >> 4;
    const int l16  = lane & 15;
    const long long base = ((long long)blockIdx.x * 8 + wave) * 32;

    if (base >= n_nodes) return;     // wave-uniform exit

    long long r0 = base + l16;       if (r0 >= n_nodes) r0 = n_nodes - 1;
    long long r1 = base + 16 + l16;  if (r1 >= n_nodes) r1 = n_nodes - 1;
    const float* a0p = hid + r0 * 64;
    const float* a1p = hid + r1 * 64;

    v8f c0[3] = {};
    v8f c1[3] = {};
    for (int kb = 0; kb < 64; kb += 4) {
        const int klo = kb + half * 2;
        v2f a0 = *(const v2f*)(a0p + klo);
        v2f a1 = *(const v2f*)(a1p + klo);
        const float* bbase = sB + (klo >> 1) * 96;
#pragma unroll
        for (int nt = 0; nt < 3; nt++) {
            v2f b = *(const v2f*)(bbase + (nt * 16 + l16) * 2);
            c0[nt] = WMMA_F32(a0, b, c0[nt]);
            c1[nt] = WMMA_F32(a1, b, c1[nt]);
        }
    }

    if (base + 32 <= n_nodes) {
#pragma unroll
        for (int nt = 0; nt < 3; nt++) {
            int col = nt * 16 + l16;
            if (col < 40) {
#pragma unroll
                for (int j = 0; j < 8; j++) {
                    long long rr = base + j + half * 8;
                    h2[rr * 40 + col]        = c0[nt][j];
                    h2[(rr + 16) * 40 + col] = c1[nt][j];
                }
            }
        }
    } else {
#pragma unroll
        for (int nt = 0; nt < 3; nt++) {
            int col = nt * 16 + l16;
            if (col < 40) {
#pragma unroll
                for (int j = 0; j < 8; j++) {
                    long long rr = base + j + half * 8;
                    if (rr < n_nodes)      h2[rr * 40 + col]        = c0[nt][j];
                    if (rr + 16 < n_nodes) h2[(rr + 16) * 40 + col] = c1[nt][j];
                }
            }
        }
    }
}

// ---------------------------------------------------------------------------
// out init: out[i,j] = b2[j]  (40 cols)
// ---------------------------------------------------------------------------
__global__ void out_init_kernel(float* __restrict__ out, const float* __restrict__ b2,
                                long long total) {
    long long t = (long long)blockIdx.x * blockDim.x + threadIdx.x;
    if (t < total) out[t] = b2[t % 40];
}

// ---------------------------------------------------------------------------
// Edge scatter, layer 2: out[dst] += h2[src] * norm (40 floats per edge,
// 8 lanes/edge, 5 floats per lane)
// ---------------------------------------------------------------------------
__global__ void scatter40_kernel(const float* __restrict__ h,
                                 const long long* __restrict__ src,
                                 const long long* __restrict__ dst,
                                 const float* __restrict__ dinv,
                                 float* __restrict__ out,
                                 long long E, int n_nodes) {
    long long t = (long long)blockIdx.x * blockDim.x + threadIdx.x;
    long long e = t >> 3;
    if (e >= E + n_nodes) return;
    int seg = (int)(t & 7) * 5;

    long long s, d;
    float w;
    if (e < E) {
        s = src[e]; d = dst[e];
        w = dinv[s] * dinv[d];
    } else {
        s = d = e - E;
        float di = dinv[s];
        w = di * di;
    }
    const float* hv = h + s * 40 + seg;
    float* o = out + d * 40 + seg;
#pragma unroll
    for (int j = 0; j < 5; j++) atomicAdd(o + j, hv[j] * w);
}

// ---------------------------------------------------------------------------
// Launch
// ---------------------------------------------------------------------------
extern "C" void kernel_launch(void* const* d_in, const int* in_sizes, int n_in,
                              void* d_out, int out_size, void* d_ws, size_t ws_size,
                              hipStream_t stream) {
    const float*     x     = (const float*)d_in[0];
    const long long* edges = (const long long*)d_in[1];   // [2, E] int64
    const float*     W1    = (const float*)d_in[2];
    const float*     b1    = (const float*)d_in[3];
    const float*     W2    = (const float*)d_in[4];
    const float*     b2    = (const float*)d_in[5];
    float*           out   = (float*)d_out;

    const int       n_nodes = in_sizes[0] / 128;
    const long long E       = (long long)in_sizes[1] / 2;
    const long long* src = edges;
    const long long* dst = edges + E;

    // workspace: dinv[N] | bufA[N*64] | bufB[N*64]
    float* dinv = (float*)d_ws;
    float* bufA = dinv + n_nodes;
    float* bufB = bufA + (size_t)n_nodes * 64;

    const int B = 256;

    // 1) degree -> dinv
    deg_init_kernel<<<(n_nodes + B - 1) / B, B, 0, stream>>>(dinv, n_nodes);
    deg_accum_kernel<<<(int)((E + B - 1) / B), B, 0, stream>>>(dst, dinv, E);
    deg_rsqrt_kernel<<<(n_nodes + B - 1) / B, B, 0, stream>>>(dinv, n_nodes);

    // 2) zero the layer-1 aggregation buffer
    hipMemsetAsync(bufB, 0, (size_t)n_nodes * 64 * sizeof(float), stream);

    // 3) GEMM1 (WMMA): bufA = x @ W1   (32 rows per wave, 8 waves per block)
    const int tiles32 = (n_nodes + 31) / 32;
    const int gblks   = (tiles32 + 7) / 8;
    gemm1_kernel<<<gblks, B, 0, stream>>>(x, W1, bufA, n_nodes);

    // 4) scatter layer 1: bufB[dst] += bufA[src] * norm
    {
        long long total = (E + n_nodes) * 16;
        scatter64_kernel<<<(int)((total + B - 1) / B), B, 0, stream>>>(
            bufA, src, dst, dinv, bufB, E, n_nodes);
    }

    // 5) hidden = relu(bufB + b1), in place
    {
        long long total = (long long)n_nodes * 64;
        bias_relu_kernel<<<(int)((total + B - 1) / B), B, 0, stream>>>(bufB, b1, total);
    }

    // 6) GEMM2 (WMMA): bufA (stride 40) = hidden @ W2
    gemm2_kernel<<<gblks, B, 0, stream>>>(bufB, W2, bufA, n_nodes);

    // 7) out = b2 broadcast, then scatter layer 2 into out
    {
        long long total = (long long)n_nodes * 40;
        out_init_kernel<<<(int)((total + B - 1) / B), B, 0, stream>>>(out, b2, total);
    }
    {
        long long total = (E + n_nodes) * 8;
        scatter40_kernel<<<(int)((total + B - 1) / B), B, 0, stream>>>(
            bufA, src, dst, dinv, out, E, n_nodes);
    }
}